// DecoderRNN_52441550684336
// MI455X (gfx1250) — compile-verified
//
#include <hip/hip_runtime.h>
#include <hip/hip_bf16.h>
#include <math.h>

#define Hs 1024
#define Vs 50257
#define Ls 64

typedef float v2f __attribute__((ext_vector_type(2)));
typedef float v8f __attribute__((ext_vector_type(8)));

// ---------------------------------------------------------------------------
// Kernel A: embedding gather + additive attention + softmax + context vector.
// One workgroup (256 threads = 8 waves). Writes:
//   ws_embed   = emb[input]                (H floats, at ws+0)
//   ws_applied = attn_weights @ enc        (H floats, at ws+H  -> contiguous cat)
//   out_attnw  = attn_weights              (L floats, into d_out tail)
// ---------------------------------------------------------------------------
__global__ void __launch_bounds__(256)
attn_kernel(const int* __restrict__ input, const float* __restrict__ hidden,
            const float* __restrict__ enc, const float* __restrict__ emb,
            const float* __restrict__ attn_W, const float* __restrict__ attn_b,
            float* __restrict__ ws_embed, float* __restrict__ ws_applied,
            float* __restrict__ out_attnw) {
    __shared__ float cat[2 * Hs];
    __shared__ float wgt[Ls];

    const int tid = threadIdx.x;
    const int idx = input[0];

    for (int i = tid; i < Hs; i += 256) {
        float e = emb[(long)idx * Hs + i];
        cat[i] = e;
        ws_embed[i] = e;
        cat[Hs + i] = hidden[i];
    }
    __syncthreads();

    // 64 logit rows: 8 waves x 8 rows, lane-strided dot over 2H with shuffle reduce
    __shared__ float logits[Ls];
    const int lane = tid & 31;
    const int wv = tid >> 5;
    for (int r8 = 0; r8 < 8; ++r8) {
        const int row = wv * 8 + r8;
        const float* wr = attn_W + (long)row * (2 * Hs);
        float s = 0.f;
        for (int k = lane; k < 2 * Hs; k += 32) s += wr[k] * cat[k];
        for (int off = 16; off > 0; off >>= 1) s += __shfl_xor(s, off, 32);
        if (lane == 0) logits[row] = s + attn_b[row];
    }
    __syncthreads();

    // softmax over 64 by wave 0 (2 values per lane)
    if (wv == 0) {
        float a0 = logits[lane], a1 = logits[lane + 32];
        float m = fmaxf(a0, a1);
        for (int off = 16; off > 0; off >>= 1) m = fmaxf(m, __shfl_xor(m, off, 32));
        float e0 = __expf(a0 - m), e1 = __expf(a1 - m);
        float s = e0 + e1;
        for (int off = 16; off > 0; off >>= 1) s += __shfl_xor(s, off, 32);
        float w0 = e0 / s, w1 = e1 / s;
        wgt[lane] = w0; wgt[lane + 32] = w1;
        out_attnw[lane] = w0; out_attnw[lane + 32] = w1;
    }
    __syncthreads();

    // attn_applied[c] = sum_l wgt[l] * enc[l][c]; coalesced over c
    for (int c = tid; c < Hs; c += 256) {
        float s = 0.f;
        #pragma unroll
        for (int l = 0; l < Ls; ++l) s += wgt[l] * enc[l * Hs + c];
        ws_applied[c] = s;
    }
}

// ---------------------------------------------------------------------------
// WMMA fp32 GEMV: y[r] = act( W[r,:] . x + bias[r] ), one wave per 16 rows.
// A-tile = 16x4 of W  (lane L -> row L&15, K-pair (L>>4)*2, per ISA layout)
// B-tile = x chunk broadcast across all 16 N columns -> every D column == y.
// Two accumulators (k step 8) to break the WMMA C-chain dependency.
// Wave-uniform control flow only, so EXEC stays all-ones around V_WMMA.
// ---------------------------------------------------------------------------
__global__ void __launch_bounds__(256)
gemv_wmma(const float* __restrict__ W, const float* __restrict__ x,
          const float* __restrict__ bias, float* __restrict__ y,
          int R, int Kd, int relu) {
    const int lane = threadIdx.x & 31;
    const int wave = (blockIdx.x * blockDim.x + threadIdx.x) >> 5;
    const int rbase = wave * 16;
    const int tiles = (R + 15) >> 4;
    if (wave >= tiles) return;                 // wave-uniform exit

    long row = rbase + (lane & 15);
    if (row >= R) row = R - 1;                 // per-lane clamp via cndmask (no EXEC change)
    const int kc = (lane >> 4) << 1;           // 0 or 2
    const float* __restrict__ wp = W + row * (long)Kd + kc;
    const float* __restrict__ xp = x + kc;

    v8f c0 = {}; v8f c1 = {};
    #pragma unroll 4
    for (int k = 0; k < Kd; k += 8) {
        v2f a0 = { wp[k],     wp[k + 1] };
        v2f b0 = { xp[k],     xp[k + 1] };
        v2f a1 = { wp[k + 4], wp[k + 5] };
        v2f b1 = { xp[k + 4], xp[k + 5] };
        c0 = __builtin_amdgcn_wmma_f32_16x16x4_f32(false, a0, false, b0,
                                                   (short)0, c0, false, false);
        c1 = __builtin_amdgcn_wmma_f32_16x16x4_f32(false, a1, false, b1,
                                                   (short)0, c1, false, false);
    }

    // D layout: VGPR j, lane 0 -> D[j,0]; lane 16 -> D[8+j,0]. Columns identical.
    if ((lane & 15) == 0) {
        const int moff = (lane >> 4) * 8;
        #pragma unroll
        for (int j = 0; j < 8; ++j) {
            long r = (long)rbase + moff + j;
            if (r < R) {
                float v = c0[j] + c1[j] + bias[r];
                if (relu) v = fmaxf(v, 0.f);
                y[r] = v;
            }
        }
    }
}

// ---------------------------------------------------------------------------
// GRU gate math (PyTorch r,z,n ordering). Writes h_new to ws and to d_out.
// ---------------------------------------------------------------------------
__global__ void __launch_bounds__(256)
gru_gate_kernel(const float* __restrict__ gi, const float* __restrict__ gh,
                const float* __restrict__ h0, float* __restrict__ hnew_ws,
                float* __restrict__ hnew_out) {
    const int i = blockIdx.x * blockDim.x + threadIdx.x;
    if (i < Hs) {
        float r = 1.f / (1.f + __expf(-(gi[i] + gh[i])));
        float z = 1.f / (1.f + __expf(-(gi[Hs + i] + gh[Hs + i])));
        float n = tanhf(gi[2 * Hs + i] + r * gh[2 * Hs + i]);
        float h = (1.f - z) * n + z * h0[i];
        hnew_ws[i] = h;
        hnew_out[i] = h;
    }
}

// ---------------------------------------------------------------------------
// log-softmax pass 1: global max and log-sum-exp over V logits (one block).
// ---------------------------------------------------------------------------
__global__ void __launch_bounds__(1024)
lsm_reduce(const float* __restrict__ logits, float* __restrict__ stats) {
    __shared__ float sm[1024];
    const int tid = threadIdx.x;
    float m = -INFINITY;
    for (int i = tid; i < Vs; i += 1024) m = fmaxf(m, logits[i]);
    sm[tid] = m; __syncthreads();
    for (int s = 512; s > 0; s >>= 1) {
        if (tid < s) sm[tid] = fmaxf(sm[tid], sm[tid + s]);
        __syncthreads();
    }
    const float gmax = sm[0];
    __syncthreads();
    float acc = 0.f;
    for (int i = tid; i < Vs; i += 1024) acc += __expf(logits[i] - gmax);
    sm[tid] = acc; __syncthreads();
    for (int s = 512; s > 0; s >>= 1) {
        if (tid < s) sm[tid] += sm[tid + s];
        __syncthreads();
    }
    if (tid == 0) { stats[0] = gmax; stats[1] = __logf(sm[0]); }
}

__global__ void __launch_bounds__(256)
lsm_final(const float* __restrict__ logits, const float* __restrict__ stats,
          float* __restrict__ out) {
    const int i = blockIdx.x * blockDim.x + threadIdx.x;
    if (i < Vs) out[i] = logits[i] - stats[0] - stats[1];
}

// ---------------------------------------------------------------------------
extern "C" void kernel_launch(void* const* d_in, const int* in_sizes, int n_in,
                              void* d_out, int out_size, void* d_ws, size_t ws_size,
                              hipStream_t stream) {
    const int*   input  = (const int*)  d_in[0];
    const float* hidden = (const float*)d_in[1];   // [1,1,H] -> H floats
    const float* enc    = (const float*)d_in[2];   // [L,H]
    const float* emb    = (const float*)d_in[3];   // [V,H]
    const float* attn_W = (const float*)d_in[4];   // [L,2H]
    const float* attn_b = (const float*)d_in[5];
    const float* comb_W = (const float*)d_in[6];   // [H,2H]
    const float* comb_b = (const float*)d_in[7];
    const float* w_ih   = (const float*)d_in[8];   // [3H,H]
    const float* w_hh   = (const float*)d_in[9];   // [3H,H]
    const float* b_ih   = (const float*)d_in[10];
    const float* b_hh   = (const float*)d_in[11];
    const float* out_W  = (const float*)d_in[12];  // [V,H]
    const float* out_b  = (const float*)d_in[13];

    float* ws         = (float*)d_ws;
    float* ws_embed   = ws;               // H     (embedded)
    float* ws_applied = ws + Hs;          // H     ([embedded;applied] contiguous at ws)
    float* ws_x       = ws + 2 * Hs;      // H     (relu(comb))
    float* ws_gi      = ws + 3 * Hs;      // 3H
    float* ws_gh      = ws + 6 * Hs;      // 3H
    float* ws_hnew    = ws + 9 * Hs;      // H
    float* ws_logits  = ws + 10 * Hs;     // V
    float* ws_stats   = ws + 10 * Hs + ((Vs + 15) & ~15); // 2 floats

    float* out       = (float*)d_out;     // [V] log-softmax
    float* out_h     = out + Vs;          // [H] h_new
    float* out_attnw = out + Vs + Hs;     // [L] attn weights

    // 1) embedding + attention (single WG)
    attn_kernel<<<1, 256, 0, stream>>>(input, hidden, enc, emb, attn_W, attn_b,
                                       ws_embed, ws_applied, out_attnw);

    // 2) x = relu(comb_W @ [embedded;applied] + comb_b): R=1024, K=2048 -> 64 waves
    gemv_wmma<<<8, 256, 0, stream>>>(comb_W, ws, comb_b, ws_x, Hs, 2 * Hs, 1);

    // 3) GRU gate pre-activations: R=3072, K=1024 -> 192 waves each
    gemv_wmma<<<24, 256, 0, stream>>>(w_ih, ws_x,   b_ih, ws_gi, 3 * Hs, Hs, 0);
    gemv_wmma<<<24, 256, 0, stream>>>(w_hh, hidden, b_hh, ws_gh, 3 * Hs, Hs, 0);

    // 4) gate nonlinearity -> h_new
    gru_gate_kernel<<<(Hs + 255) / 256, 256, 0, stream>>>(ws_gi, ws_gh, hidden,
                                                          ws_hnew, out_h);

    // 5) vocab logits: R=50257, K=1024 -> 3142 waves -> 393 WGs (streams 206 MB)
    gemv_wmma<<<(((Vs + 15) / 16) + 7) / 8, 256, 0, stream>>>(out_W, ws_hnew, out_b,
                                                              ws_logits, Vs, Hs, 0);

    // 6) log-softmax
    lsm_reduce<<<1, 1024, 0, stream>>>(ws_logits, ws_stats);
    lsm_final<<<(Vs + 255) / 256, 256, 0, stream>>>(ws_logits, ws_stats, out);
}